// CPHashLayer_86732569576121
// MI455X (gfx1250) — compile-verified
//
#include <hip/hip_runtime.h>
#include <cstddef>
#include <cstdint>

// ---------------------------------------------------------------------------
// CP-ALS (rank 32) + 2-layer MLP hash head for gfx1250 (MI455X).
// - All GEMM-shaped work uses V_WMMA_F32_16X16X4_F32 (fp32 WMMA, wave32).
// - GEMM1 streams W1 through the Tensor Data Mover (TENSOR_LOAD_TO_LDS,
//   TENSORcnt double-buffered) so the 597MB weight stream rides the DMA path
//   while wave32s issue WMMAs from LDS B-fragments.
// ---------------------------------------------------------------------------

typedef __attribute__((ext_vector_type(2))) float v2f;
typedef __attribute__((ext_vector_type(8))) float v8f;

#define BATCH 64
#define CD    512
#define HD    14
#define WD    14
#define HW    (HD * WD)            // 196
#define R     32
#define NIT   10
#define FK    (R * (CD + HD + WD)) // 17280
#define HID   (FK / 2)             // 8640
#define HL    128
#define KCH   64                   // k-chunk staged per TDM transfer

// fp32 WMMA: D(16x16) = A(16x4) * B(4x16) + C.
__device__ __forceinline__ v8f wmma_f32(v2f a, v2f b, v8f c) {
  return __builtin_amdgcn_wmma_f32_16x16x4_f32(false, a, false, b, (short)0, c,
                                               false, false);
}

#if defined(__has_builtin)
#if __has_builtin(__builtin_amdgcn_tensor_load_to_lds) && \
    __has_builtin(__builtin_amdgcn_s_wait_tensorcnt)
#define USE_TDM 1
#endif
#endif
#ifndef USE_TDM
#define USE_TDM 0
#endif

#if USE_TDM
typedef unsigned int tdm_g0_t __attribute__((ext_vector_type(4)));
typedef int          tdm_g1_t __attribute__((ext_vector_type(8)));
typedef int          tdm_g2_t __attribute__((ext_vector_type(4)));
typedef int          tdm_g4_t __attribute__((ext_vector_type(8)));

// Issue a TDM load of a [16 x KCH] fp32 tile of W1 (row stride FK) into LDS.
__device__ __forceinline__ void tdm_load_w1_panel(const float* gtile,
                                                  unsigned lds_byte_addr) {
  unsigned long long ga = (unsigned long long)(uintptr_t)gtile;
  tdm_g0_t g0;
  g0[0] = 1u;                                   // count=1, user descriptor
  g0[1] = lds_byte_addr;                        // lds_addr [63:32]
  g0[2] = (unsigned)ga;                         // global_addr [95:64]
  g0[3] = (unsigned)((ga >> 32) & 0x01FFFFFFull) | (2u << 30); // addr hi | type=2
  tdm_g1_t g1;
  g1[0] = (int)(2u << 16);                      // workgroup_mask=0, data_size=4B
  g1[1] = (int)((FK & 0xFFFFu) << 16);          // tensor_dim0 lo16 in [63:48]
  g1[2] = (int)(((HID & 0xFFFFu) << 16) | ((FK >> 16) & 0xFFFFu));  // dim1 lo | dim0 hi
  g1[3] = (int)(((KCH & 0xFFFFu) << 16) | ((HID >> 16) & 0xFFFFu)); // tile_dim0 | dim1 hi
  g1[4] = 16;                                   // tile_dim1=16 rows, tile_dim2=0
  g1[5] = (int)FK;                              // tensor_dim0_stride lo32
  g1[6] = 0;                                    // stride hi | dim1_stride lo
  g1[7] = 0;
  tdm_g2_t gz4 = {0, 0, 0, 0};                  // 2-D tensor: groups 2/3 unused
  tdm_g4_t gz8 = {0, 0, 0, 0, 0, 0, 0, 0};      // extra group (clang-23 6-arg form)
  __builtin_amdgcn_tensor_load_to_lds(g0, g1, gz4, gz4, gz8, 0);
}
#endif

// ---------------- deterministic pseudo-normal init ------------------------
__device__ __forceinline__ unsigned hashu(unsigned x) {
  x ^= x >> 16; x *= 0x7feb352dU;
  x ^= x >> 15; x *= 0x846ca68bU;
  x ^= x >> 16; return x;
}

__device__ __forceinline__ float rng_normal(unsigned b, unsigned i) {
  unsigned s1 = hashu((b + 1u) * 0x9E3779B9u ^ (i * 2u + 1u));
  unsigned s2 = hashu(s1 ^ 0x85ebca6bu ^ i);
  float u1 = (float)((s1 >> 8) + 1u) * (1.0f / 16777217.0f);
  float u2 = (float)(s2 >> 8) * (1.0f / 16777216.0f);
  return sqrtf(-2.0f * __logf(u1)) * __cosf(6.28318530718f * u2);
}

// ------------- block-cooperative 32x32 inverse of G1 .* G2 ----------------
__device__ void invert32(const float* __restrict__ G1,
                         const float* __restrict__ G2,
                         float* __restrict__ aug, float* __restrict__ P,
                         int tid) {
  for (int e = tid; e < 32 * 64; e += 256) {
    int r = e >> 6, c = e & 63;
    aug[e] = (c < 32) ? G1[r * 32 + c] * G2[r * 32 + c]
                      : ((c - 32) == r ? 1.0f : 0.0f);
  }
  __syncthreads();
  int r  = tid >> 3;
  int c0 = tid & 7;
  for (int col = 0; col < 32; ++col) {
    float piv = aug[col * 64 + col];
    float ip  = 1.0f / (piv + (piv >= 0.0f ? 1e-20f : -1e-20f));
    __syncthreads();
    if (tid < 64) aug[col * 64 + tid] *= ip;
    __syncthreads();
    float f = (r != col) ? aug[r * 64 + col] : 0.0f;
    __syncthreads();
    if (r != col)
      for (int c = c0; c < 64; c += 8)
        aug[r * 64 + c] = fmaf(-f, aug[col * 64 + c], aug[r * 64 + c]);
    __syncthreads();
  }
  for (int e = tid; e < 32 * 32; e += 256)
    P[e] = aug[(e >> 5) * 64 + 32 + (e & 31)];
  __syncthreads();
}

// --------------------------- ALS kernel -----------------------------------
__global__ __launch_bounds__(256) void als_kernel(
    const float* __restrict__ x, float* __restrict__ wsA,
    float* __restrict__ wsMa, float* __restrict__ wsBf,
    float* __restrict__ wsCf) {
  __shared__ float sB[HD * R], sC[HD * R];
  __shared__ float sKA[HW * R];
  __shared__ float sGA[R * R], sGB[R * R], sGC[R * R];
  __shared__ float sP[R * R];
  __shared__ float sAug[R * 2 * R];
  __shared__ float sMB[HD * R], sMC[HD * R];

  const int b    = blockIdx.x;
  const int tid  = threadIdx.x;
  const int wave = tid >> 5;
  const int lane = tid & 31;
  const int half = lane >> 4;
  const int l16  = lane & 15;

  const float* X  = x + (size_t)b * CD * HW;
  float*       A  = wsA + (size_t)b * CD * R;
  float*       Ma = wsMa + (size_t)b * CD * R;

  for (int i = tid; i < CD * R; i += 256) A[i] = rng_normal(b, i);
  for (int i = tid; i < HD * R; i += 256) {
    sB[i] = rng_normal(b, 0x10000u + i);
    sC[i] = rng_normal(b, 0x20000u + i);
  }
  __syncthreads();

  for (int it = 0; it < NIT; ++it) {
    // (1) Gram_B, Gram_C
    for (int e = tid; e < R * R; e += 256) {
      int i = e >> 5, j = e & 31;
      float gb = 0.0f, gc = 0.0f;
#pragma unroll
      for (int h = 0; h < HD; ++h) {
        gb = fmaf(sB[h * R + i], sB[h * R + j], gb);
        gc = fmaf(sC[h * R + i], sC[h * R + j], gc);
      }
      sGB[e] = gb; sGC[e] = gc;
    }
    __syncthreads();

    // (2) P_A = inv(Gram_B .* Gram_C)
    invert32(sGB, sGC, sAug, sP, tid);

    // (3) K_A = KhatriRao(B, C)
    for (int e = tid; e < HW * R; e += 256) {
      int hw = e >> 5, r = e & 31;
      sKA[e] = sB[(hw / WD) * R + r] * sC[(hw % WD) * R + r];
    }
    __syncthreads();

    // (4) M_A = X0 @ K_A   (WMMA: 512x32x196)
    for (int t = wave; t < 32 * 2; t += 8) {
      int mt = t >> 1, nt = t & 1;
      v8f acc = {0.f, 0.f, 0.f, 0.f, 0.f, 0.f, 0.f, 0.f};
      const float* xr = X + (size_t)(mt * 16 + l16) * HW;
      for (int kk = 0; kk < HW; kk += 4) {
        int kb = kk + half * 2;
        v2f a = {xr[kb], xr[kb + 1]};
        v2f bb = {sKA[kb * R + nt * 16 + l16],
                  sKA[(kb + 1) * R + nt * 16 + l16]};
        acc = wmma_f32(a, bb, acc);
      }
#pragma unroll
      for (int i = 0; i < 8; ++i)
        Ma[(size_t)(mt * 16 + i + half * 8) * R + nt * 16 + l16] = acc[i];
    }
    __syncthreads();

    // (5) A = M_A @ P_A   (WMMA: 512x32x32)
    for (int t = wave; t < 32 * 2; t += 8) {
      int mt = t >> 1, nt = t & 1;
      v8f acc = {0.f, 0.f, 0.f, 0.f, 0.f, 0.f, 0.f, 0.f};
      const float* mr = Ma + (size_t)(mt * 16 + l16) * R;
      for (int kk = 0; kk < R; kk += 4) {
        int kb = kk + half * 2;
        v2f a = {mr[kb], mr[kb + 1]};
        v2f bb = {sP[kb * R + nt * 16 + l16],
                  sP[(kb + 1) * R + nt * 16 + l16]};
        acc = wmma_f32(a, bb, acc);
      }
#pragma unroll
      for (int i = 0; i < 8; ++i)
        A[(size_t)(mt * 16 + i + half * 8) * R + nt * 16 + l16] = acc[i];
    }
    __syncthreads();

    // (6) Gram_A = A^T A   (WMMA: 32x32x512), waves 0..3
    if (wave < 4) {
      int mt = wave >> 1, nt = wave & 1;
      v8f acc = {0.f, 0.f, 0.f, 0.f, 0.f, 0.f, 0.f, 0.f};
      for (int kk = 0; kk < CD; kk += 4) {
        int kb = kk + half * 2;
        v2f a = {A[(size_t)kb * R + mt * 16 + l16],
                 A[(size_t)(kb + 1) * R + mt * 16 + l16]};
        v2f bb = {A[(size_t)kb * R + nt * 16 + l16],
                  A[(size_t)(kb + 1) * R + nt * 16 + l16]};
        acc = wmma_f32(a, bb, acc);
      }
#pragma unroll
      for (int i = 0; i < 8; ++i)
        sGA[(mt * 16 + i + half * 8) * R + nt * 16 + l16] = acc[i];
    }
    __syncthreads();

    // (7) P_B = inv(Gram_A .* Gram_C)
    invert32(sGA, sGC, sAug, sP, tid);

    // (8) M_B[h,r] = sum_{c,w} X[c,h,w] A[c,r] C[w,r]
    for (int e = tid; e < HD * R; e += 256) sMB[e] = 0.0f;
    __syncthreads();
    {
      float part[HD];
#pragma unroll
      for (int h = 0; h < HD; ++h) part[h] = 0.0f;
      for (int c = wave; c < CD; c += 8) {
        float a = A[(size_t)c * R + lane];
        const float* xc = X + (size_t)c * HW;
#pragma unroll
        for (int h = 0; h < HD; ++h) {
          float s = 0.0f;
#pragma unroll
          for (int w = 0; w < WD; ++w)
            s = fmaf(xc[h * WD + w], sC[w * R + lane], s);
          part[h] = fmaf(a, s, part[h]);
        }
      }
#pragma unroll
      for (int h = 0; h < HD; ++h) atomicAdd(&sMB[h * R + lane], part[h]);
    }
    __syncthreads();

    // (9) B = M_B @ P_B
    for (int e = tid; e < HD * R; e += 256) {
      int h = e >> 5, r = e & 31;
      float s = 0.0f;
#pragma unroll
      for (int k = 0; k < R; ++k) s = fmaf(sMB[h * R + k], sP[k * R + r], s);
      sB[e] = s;
    }
    __syncthreads();

    // (10) Gram_B (new B)
    for (int e = tid; e < R * R; e += 256) {
      int i = e >> 5, j = e & 31;
      float gb = 0.0f;
#pragma unroll
      for (int h = 0; h < HD; ++h)
        gb = fmaf(sB[h * R + i], sB[h * R + j], gb);
      sGB[e] = gb;
    }
    __syncthreads();

    // (11) P_C = inv(Gram_A .* Gram_B)
    invert32(sGA, sGB, sAug, sP, tid);

    // (12) M_C[w,r] = sum_{c,h} X[c,h,w] A[c,r] B[h,r]
    for (int e = tid; e < HD * R; e += 256) sMC[e] = 0.0f;
    __syncthreads();
    {
      float part[WD];
#pragma unroll
      for (int w = 0; w < WD; ++w) part[w] = 0.0f;
      for (int c = wave; c < CD; c += 8) {
        float a = A[(size_t)c * R + lane];
        const float* xc = X + (size_t)c * HW;
#pragma unroll
        for (int h = 0; h < HD; ++h) {
          float coef = a * sB[h * R + lane];
#pragma unroll
          for (int w = 0; w < WD; ++w)
            part[w] = fmaf(xc[h * WD + w], coef, part[w]);
        }
      }
#pragma unroll
      for (int w = 0; w < WD; ++w) atomicAdd(&sMC[w * R + lane], part[w]);
    }
    __syncthreads();

    // (13) C = M_C @ P_C
    for (int e = tid; e < HD * R; e += 256) {
      int w = e >> 5, r = e & 31;
      float s = 0.0f;
#pragma unroll
      for (int k = 0; k < R; ++k) s = fmaf(sMC[w * R + k], sP[k * R + r], s);
      sC[e] = s;
    }
    __syncthreads();
  }

  for (int e = tid; e < HD * R; e += 256) {
    wsBf[(size_t)b * HD * R + e] = sB[e];
    wsCf[(size_t)b * HD * R + e] = sC[e];
  }
}

// ------------------ pack factor_batch [64, 17280] --------------------------
__global__ __launch_bounds__(256) void pack_kernel(
    const float* __restrict__ A, const float* __restrict__ Bf,
    const float* __restrict__ Cf, float* __restrict__ F) {
  int b = blockIdx.x;
  for (int k = threadIdx.x; k < FK; k += 256) {
    float v;
    if (k < CD * R)               v = A[(size_t)b * CD * R + k];
    else if (k < CD * R + HD * R) v = Bf[(size_t)b * HD * R + (k - CD * R)];
    else                          v = Cf[(size_t)b * HD * R + (k - CD * R - HD * R)];
    F[(size_t)b * FK + k] = v;
  }
}

// -------- GEMM1: H1 = relu(F @ W1^T + b1)   M=64, N=8640, K=17280 ----------
// One block per 16-col N tile; 4 waves cover the 4 M tiles.  W1 panels
// (16 rows x KCH k) are DMA'd into double-buffered LDS by the TDM; wave 0
// owns the TENSORcnt pipeline, workgroup barriers publish each panel.
__global__ __launch_bounds__(128) void gemm1_kernel(
    const float* __restrict__ F, const float* __restrict__ W1,
    const float* __restrict__ b1, float* __restrict__ H1) {
  __shared__ float sW[2][16 * KCH];                // 2 x 4KB panels

  const int ntile = blockIdx.x;                    // 0..539
  const int mtile = threadIdx.x >> 5;              // wave id 0..3
  const int lane  = threadIdx.x & 31;
  const int half  = lane >> 4;
  const int l16   = lane & 15;
  const float* frow  = F  + (size_t)(mtile * 16 + l16) * FK;
  const float* wbase = W1 + (size_t)ntile * 16 * FK;

  v8f acc = {0.f, 0.f, 0.f, 0.f, 0.f, 0.f, 0.f, 0.f};
  const int NCH = FK / KCH;                        // 270 chunks

#if USE_TDM
  if (mtile == 0)
    tdm_load_w1_panel(wbase, (unsigned)(uintptr_t)(const void*)&sW[0][0]);
#endif
  for (int ch = 0; ch < NCH; ++ch) {
#if USE_TDM
    if (mtile == 0) {
      if (ch + 1 < NCH) {
        tdm_load_w1_panel(wbase + (size_t)(ch + 1) * KCH,
                          (unsigned)(uintptr_t)(const void*)&sW[(ch + 1) & 1][0]);
        __builtin_amdgcn_s_wait_tensorcnt(1);      // chunk ch has landed
      } else {
        __builtin_amdgcn_s_wait_tensorcnt(0);
      }
    }
#else
    // fallback: cooperative load of the panel
    for (int t = threadIdx.x; t < 16 * KCH; t += 128)
      sW[ch & 1][t] = wbase[(size_t)(t / KCH) * FK + ch * KCH + (t % KCH)];
#endif
    __syncthreads();                               // publish panel ch
    const float* wp    = &sW[ch & 1][0];
    const float* fbase = frow + (size_t)ch * KCH;
    __builtin_prefetch(fbase + KCH, 0, 1);         // next F chunk
    for (int kk = 0; kk < KCH; kk += 4) {
      int kb = kk + half * 2;
      v2f a  = {fbase[kb], fbase[kb + 1]};
      v2f bb = {wp[l16 * KCH + kb], wp[l16 * KCH + kb + 1]};
      acc = wmma_f32(a, bb, acc);
    }
    __syncthreads();                               // done reading buf ch&1
  }
#pragma unroll
  for (int i = 0; i < 8; ++i) {
    int m = mtile * 16 + i + half * 8;
    int n = ntile * 16 + l16;
    float v = acc[i] + b1[n];
    H1[(size_t)m * HID + n] = v > 0.0f ? v : 0.0f;
  }
}

// -------- GEMM2: out = H1 @ W2^T + b2 ; binary = sign(out) -----------------
__global__ __launch_bounds__(128) void gemm2_kernel(
    const float* __restrict__ H1, const float* __restrict__ W2,
    const float* __restrict__ b2, float* __restrict__ out) {
  const int ntile = blockIdx.x;                    // 0..7
  const int mtile = threadIdx.x >> 5;
  const int lane  = threadIdx.x & 31;
  const int half  = lane >> 4;
  const int l16   = lane & 15;
  const float* hrow = H1 + (size_t)(mtile * 16 + l16) * HID;
  const float* wrow = W2 + (size_t)(ntile * 16 + l16) * HID;
  v8f acc = {0.f, 0.f, 0.f, 0.f, 0.f, 0.f, 0.f, 0.f};
  for (int kk = 0; kk < HID; kk += 4) {
    int kb = kk + half * 2;
    v2f a  = {hrow[kb], hrow[kb + 1]};
    v2f bb = {wrow[kb], wrow[kb + 1]};
    acc = wmma_f32(a, bb, acc);
  }
#pragma unroll
  for (int i = 0; i < 8; ++i) {
    int m = mtile * 16 + i + half * 8;
    int n = ntile * 16 + l16;
    float ch = acc[i] + b2[n];
    out[(size_t)BATCH * HL + (size_t)m * HL + n] = ch;
    out[(size_t)m * HL + n] = (ch > 0.0f) ? 1.0f : (ch < 0.0f ? -1.0f : 0.0f);
  }
}

// ---------------------------------------------------------------------------
extern "C" void kernel_launch(void* const* d_in, const int* in_sizes, int n_in,
                              void* d_out, int out_size, void* d_ws,
                              size_t ws_size, hipStream_t stream) {
  (void)in_sizes; (void)n_in; (void)out_size; (void)ws_size;
  const float* x  = (const float*)d_in[0];
  const float* W1 = (const float*)d_in[1];
  const float* b1 = (const float*)d_in[2];
  const float* W2 = (const float*)d_in[3];
  const float* b2 = (const float*)d_in[4];
  float* out = (float*)d_out;

  float* ws   = (float*)d_ws;
  float* wsA  = ws;                                  // 64*512*32
  float* wsMa = wsA + (size_t)BATCH * CD * R;
  float* wsBf = wsMa + (size_t)BATCH * CD * R;
  float* wsCf = wsBf + (size_t)BATCH * HD * R;
  float* wsF  = wsCf + (size_t)BATCH * HD * R;       // 64*17280
  float* wsH1 = wsF + (size_t)BATCH * FK;            // 64*8640

  als_kernel<<<BATCH, 256, 0, stream>>>(x, wsA, wsMa, wsBf, wsCf);
  pack_kernel<<<BATCH, 256, 0, stream>>>(wsA, wsBf, wsCf, wsF);
  gemm1_kernel<<<HID / 16, 128, 0, stream>>>(wsF, W1, b1, wsH1);
  gemm2_kernel<<<HL / 16, 128, 0, stream>>>(wsH1, W2, b2, out);
}